// GraphNeuralNetwork_49203145343286
// MI455X (gfx1250) — compile-verified
//
#include <hip/hip_runtime.h>

typedef float v2f __attribute__((ext_vector_type(2)));
typedef float v8f __attribute__((ext_vector_type(8)));
typedef unsigned int u32x4 __attribute__((ext_vector_type(4)));
typedef int i32x4 __attribute__((ext_vector_type(4)));
typedef int i32x8 __attribute__((ext_vector_type(8)));

#define GCN_NODES 50000
#define GCN_EDGES 800000
#define GCN_INF   128
#define GCN_HID   128
#define GCN_OUTF  64

#define MBLK   80            // rows per block (50000 = 625 * 80)
#define MSUB   (MBLK / 16)   // 5 WMMA M-subtiles per wave
#define LDS_PITCH 132        // 128 + 4 dword pad per row (TDM pad feature)

// ---------------------------------------------------------------------------
// Degree / normalization kernels
// ---------------------------------------------------------------------------

__global__ void gcn_init_deg(float* __restrict__ deg, int n) {
  int i = blockIdx.x * blockDim.x + threadIdx.x;
  if (i < n) deg[i] = 1.0f;  // self-loop contributes 1 to every node's degree
}

__global__ void gcn_count_deg(const long long* __restrict__ dst,
                              float* __restrict__ deg, int e) {
  int i = blockIdx.x * blockDim.x + threadIdx.x;
  if (i < e) atomicAdd(&deg[(int)dst[i]], 1.0f);
}

__global__ void gcn_rsqrt_deg(float* __restrict__ deg, int n) {
  int i = blockIdx.x * blockDim.x + threadIdx.x;
  if (i < n) {
    float d = deg[i];
    deg[i] = (d > 0.0f) ? rsqrtf(d) : 0.0f;  // becomes dinv in place
  }
}

// ---------------------------------------------------------------------------
// Dense transform H = A(MxK) @ W(KxN), f32, V_WMMA_F32_16X16X4_F32.
//
// Block tile: MBLK(=80) rows x N cols. The 80xK f32 A tile is DMA'd into LDS
// once per block by the Tensor Data Mover (wave 0 issues TENSOR_LOAD_TO_LDS,
// waits on TENSORcnt, then a workgroup barrier). TDM inserts 4 dwords of LDS
// padding every 128 dwords (pad_interval=6, pad_amount=3) giving row pitch
// 132 so half-wave A reads hit 16 distinct bank pairs.
//
// Wave w owns column tile n0=16w and sweeps MSUB(=5) M-subtiles per k-step,
// so each B fragment (2 global b32 loads) feeds 5 WMMAs and A fragments come
// from LDS (ds_load_b64).
//
// VGPR layouts (ISA 7.12.2, wave32):
//   A 16x4 f32 : lanes 0-15 K={k0,k0+1}; lanes 16-31 K={k0+2,k0+3}
//   B 4x16 f32 : lane&15 = column, K = k0 + 2*(lane>>4) + vgpr
//   C/D 16x16  : vgpr v -> row base + 8*(lane>>4) + v, col n0 + (lane&15)
// ---------------------------------------------------------------------------

template <int K, int N>
__global__ void gcn_gemm_wmma(const float* __restrict__ A,
                              const float* __restrict__ W,
                              float* __restrict__ H) {
  __shared__ float ldsA[MBLK * LDS_PITCH];  // single LDS object -> offset 0

  const int wave = threadIdx.x >> 5;
  const int lane = threadIdx.x & 31;
  const int half = lane >> 4;
  const int r    = lane & 15;
  const int m0   = blockIdx.x * MBLK;
  const int n0   = wave * 16;

#if __has_builtin(__builtin_amdgcn_tensor_load_to_lds)
  if (wave == 0) {
    const unsigned long long gaddr =
        (unsigned long long)(const void*)(A + (size_t)m0 * K);

    u32x4 g0;
    g0.x = 1u;                                   // count=1, user descriptor
    g0.y = 0u;                                   // lds_addr = 0 (ldsA base)
    g0.z = (unsigned int)gaddr;                  // global_addr[31:0]
    g0.w = (unsigned int)((gaddr >> 32) & 0x01FFFFFFu) | (2u << 30);  // type=2

    i32x8 g1;
    g1[0] = (2 << 16)      // data_size = 4 bytes
          | (1 << 20)      // pad_enable
          | (6 << 22)      // pad_interval: 128 dwords
          | (3 << 25);     // pad_amount:   4 dwords
    g1[1] = (K & 0xFFFF) << 16;           // tensor_dim0 = K
    g1[2] = (MBLK & 0xFFFF) << 16;        // tensor_dim0 hi=0 | tensor_dim1=80
    g1[3] = (K & 0xFFFF) << 16;           // tensor_dim1 hi=0 | tile_dim0 = K
    g1[4] = MBLK;                         // tile_dim1 = 80 | tile_dim2 = 0
    g1[5] = K;                            // tensor_dim0_stride = K
    g1[6] = 0;                            // dim0_stride hi | dim1_stride lo
    g1[7] = 0;                            // tensor_dim1_stride hi
    i32x4 gz = {0, 0, 0, 0};              // groups 2/3: dims beyond 2D unused

#if __clang_major__ >= 23
    i32x8 gz8 = {0, 0, 0, 0, 0, 0, 0, 0};
    __builtin_amdgcn_tensor_load_to_lds(g0, g1, gz, gz, gz8, 0);
#else
    __builtin_amdgcn_tensor_load_to_lds(g0, g1, gz, gz, 0);
#endif
    __builtin_amdgcn_s_wait_tensorcnt(0);
  }
  __syncthreads();
#else
  // Fallback: cooperative global->LDS staging with padded pitch.
  for (int idx = threadIdx.x; idx < MBLK * K; idx += blockDim.x) {
    int row = idx / K, col = idx % K;
    ldsA[row * LDS_PITCH + col] = A[(size_t)(m0 + row) * K + col];
  }
  __syncthreads();
#endif

  v8f acc[MSUB];
#pragma unroll
  for (int t = 0; t < MSUB; ++t) acc[t] = (v8f){};

#pragma unroll
  for (int k0 = 0; k0 < K; k0 += 4) {
    const int ka = k0 + 2 * half;
    v2f b;
    b.x = W[(size_t)(ka + 0) * N + n0 + r];
    b.y = W[(size_t)(ka + 1) * N + n0 + r];
#pragma unroll
    for (int t = 0; t < MSUB; ++t) {
      const v2f a = *(const v2f*)&ldsA[(t * 16 + r) * LDS_PITCH + ka];
      acc[t] = __builtin_amdgcn_wmma_f32_16x16x4_f32(false, a, false, b,
                                                     (short)0, acc[t],
                                                     false, false);
    }
  }

#pragma unroll
  for (int t = 0; t < MSUB; ++t) {
#pragma unroll
    for (int v = 0; v < 8; ++v) {
      H[(size_t)(m0 + t * 16 + 8 * half + v) * N + n0 + r] = acc[t][v];
    }
  }
}

// ---------------------------------------------------------------------------
// Aggregation: agg = D^{-1/2}(A+I)D^{-1/2} @ h
// ---------------------------------------------------------------------------

__global__ void gcn_selfloop_init(const float* __restrict__ h,
                                  const float* __restrict__ dinv,
                                  float* __restrict__ agg, int n, int f) {
  int idx = blockIdx.x * blockDim.x + threadIdx.x;
  if (idx < n * f) {
    float d = dinv[idx / f];
    agg[idx] = h[idx] * d * d;
  }
}

template <int F>
__global__ void gcn_scatter(const long long* __restrict__ src,
                            const long long* __restrict__ dst,
                            const float* __restrict__ dinv,
                            const float* __restrict__ h,
                            float* __restrict__ agg, int e) {
  int gid  = blockIdx.x * blockDim.x + threadIdx.x;
  int edge = gid >> 5;
  int lane = gid & 31;
  if (edge >= e) return;

  const int s = (int)src[edge];
  const int d = (int)dst[edge];
  const float norm = dinv[s] * dinv[d];

  const float* __restrict__ hs = h + (size_t)s * F;
  float* __restrict__ ad = agg + (size_t)d * F;

  if constexpr (F == 128) {
    const float4 v = ((const float4*)hs)[lane];
    atomicAdd(&ad[4 * lane + 0], v.x * norm);
    atomicAdd(&ad[4 * lane + 1], v.y * norm);
    atomicAdd(&ad[4 * lane + 2], v.z * norm);
    atomicAdd(&ad[4 * lane + 3], v.w * norm);
  } else {
    const float2 v = ((const float2*)hs)[lane];
    atomicAdd(&ad[2 * lane + 0], v.x * norm);
    atomicAdd(&ad[2 * lane + 1], v.y * norm);
  }
}

__global__ void gcn_bias_act(float* __restrict__ a, const float* __restrict__ b,
                             int n, int f, int relu) {
  int idx = blockIdx.x * blockDim.x + threadIdx.x;
  if (idx < n * f) {
    float v = a[idx] + b[idx % f];
    a[idx] = relu ? fmaxf(v, 0.0f) : v;
  }
}

// ---------------------------------------------------------------------------
// Launch
// ---------------------------------------------------------------------------

extern "C" void kernel_launch(void* const* d_in, const int* in_sizes, int n_in,
                              void* d_out, int out_size, void* d_ws, size_t ws_size,
                              hipStream_t stream) {
  (void)in_sizes; (void)n_in; (void)out_size; (void)ws_size;

  const float*     x  = (const float*)d_in[0];
  const long long* ei = (const long long*)d_in[1];  // [2, E] int64
  const float*     W1 = (const float*)d_in[2];
  const float*     b1 = (const float*)d_in[3];
  const float*     W2 = (const float*)d_in[4];
  const float*     b2 = (const float*)d_in[5];
  float*           out = (float*)d_out;

  const long long* src = ei;              // row 0
  const long long* dst = ei + GCN_EDGES;  // row 1

  // Workspace layout (floats): dinv | h1 | agg1 | h2
  float* ws   = (float*)d_ws;
  float* dinv = ws;                                   // 50000 (+pad)
  float* h1   = ws + 50176;                           // 50000*128
  float* agg1 = h1 + (size_t)GCN_NODES * GCN_HID;     // 50000*128
  float* h2   = agg1 + (size_t)GCN_NODES * GCN_HID;   // 50000*64

  const int TB = 256;
  const int nodeBlocks  = (GCN_NODES + TB - 1) / TB;
  const int edgeBlocks  = (GCN_EDGES + TB - 1) / TB;
  const int edgeWaveBlk = (GCN_EDGES * 32 + TB - 1) / TB;  // one wave per edge
  const int elemBlocks1 = (GCN_NODES * GCN_HID + TB - 1) / TB;
  const int elemBlocks2 = (GCN_NODES * GCN_OUTF + TB - 1) / TB;
  const int gemmBlocks  = GCN_NODES / MBLK;                // 625

  // Normalization: deg = 1 + indegree; dinv = rsqrt(deg)
  gcn_init_deg<<<nodeBlocks, TB, 0, stream>>>(dinv, GCN_NODES);
  gcn_count_deg<<<edgeBlocks, TB, 0, stream>>>(dst, dinv, GCN_EDGES);
  gcn_rsqrt_deg<<<nodeBlocks, TB, 0, stream>>>(dinv, GCN_NODES);

  // Layer 1: h1 = x@W1 ; agg1 = norm-aggregate(h1) ; agg1 = relu(agg1 + b1)
  gcn_gemm_wmma<GCN_INF, GCN_HID>
      <<<gemmBlocks, (GCN_HID / 16) * 32, 0, stream>>>(x, W1, h1);
  gcn_selfloop_init<<<elemBlocks1, TB, 0, stream>>>(h1, dinv, agg1,
                                                    GCN_NODES, GCN_HID);
  gcn_scatter<GCN_HID><<<edgeWaveBlk, TB, 0, stream>>>(src, dst, dinv, h1,
                                                       agg1, GCN_EDGES);
  gcn_bias_act<<<elemBlocks1, TB, 0, stream>>>(agg1, b1, GCN_NODES, GCN_HID, 1);

  // Layer 2: h2 = agg1@W2 ; out = norm-aggregate(h2) + b2
  gcn_gemm_wmma<GCN_HID, GCN_OUTF>
      <<<gemmBlocks, (GCN_OUTF / 16) * 32, 0, stream>>>(agg1, W2, h2);
  gcn_selfloop_init<<<elemBlocks2, TB, 0, stream>>>(h2, dinv, out,
                                                    GCN_NODES, GCN_OUTF);
  gcn_scatter<GCN_OUTF><<<edgeWaveBlk, TB, 0, stream>>>(src, dst, dinv, h2,
                                                        out, GCN_EDGES);
  gcn_bias_act<<<elemBlocks2, TB, 0, stream>>>(out, b2, GCN_NODES, GCN_OUTF, 0);
}